// KStoNet_76175539962452
// MI455X (gfx1250) — compile-verified
//
#include <hip/hip_runtime.h>
#include <math.h>
#include <stdint.h>

typedef float v2f __attribute__((ext_vector_type(2)));
typedef float v8f __attribute__((ext_vector_type(8)));

#define GAMMA   0.1f
#define D_DIM   64
#define H0_DIM  256
#define K_CENT  50
#define H1_DIM  256

#define MT_PER_BLK   4                    // M-tiles (16 rows) per block -> 64 rows
#define ROWS_PER_BLK 64
#define CB_STRIDE    68                   // padded row stride (floats): 16B-aligned rows,
                                          // bank = (4*k + 4*c + 2*half) % 64 -> conflict-free b64
#define CB_ELEMS     (K_CENT * CB_STRIDE) // 3400 floats per staged head
#define HEAD_BYTES   (K_CENT * D_DIM * 4) // 12800 B of centers per head (contiguous)
#define HID_STRIDE   258

// ---- CDNA5 async copy: global -> LDS, tracked with ASYNCcnt -----------------
__device__ __forceinline__ void async_ld_b128(uint32_t lds_byte_off,
                                              unsigned long long gaddr) {
  asm volatile("global_load_async_to_lds_b128 %0, %1, off"
               :: "v"(lds_byte_off), "v"(gaddr) : "memory");
}
__device__ __forceinline__ void wait_async0() {
  asm volatile("s_wait_asynccnt 0x0" ::: "memory");
}

// Stage heads {2*pair, 2*pair+1} of centers into LDS (2 x 800 16B chunks).
__device__ __forceinline__ void stage_pair(unsigned long long centers_g, int pair,
                                           uint32_t lds_base, int tid) {
#pragma unroll
  for (int j = 0; j < 7; ++j) {
    int idx = tid + j * 256;              // chunk id in [0, 1600)
    if (idx < 1600) {                     // wave-uniform tail predicate
      int hl = idx / 800;                 // which of the 2 heads
      int ii = idx - hl * 800;            // chunk within head
      int r  = ii >> 4;                   // center row (0..49)
      int c4 = ii & 15;                   // 16B chunk within the 64-float row
      async_ld_b128(lds_base + (uint32_t)(hl * (CB_ELEMS * 4) +
                                          r * (CB_STRIDE * 4) + c4 * 16),
                    centers_g + (unsigned long long)(2 * pair + hl) * HEAD_BYTES +
                        (unsigned long long)ii * 16ull);
    }
  }
}

// ---------------------------------------------------------------------------
// One block = 64 batch rows, 8 waves = 4 M-tiles x 2 head-groups.
//   Phase 1: for each head (double-buffered async LDS staging of centers):
//            xc tile via f32 WMMA (B from LDS), exp, reduce, tanh -> hid LDS.
//   Phase 2: hid @ fc_w^T via WMMA (A from LDS), tanh, dot out_w -> out.
// ---------------------------------------------------------------------------
__global__ __launch_bounds__(256) void kstonet_fused_kernel(
    const float* __restrict__ x,      const float* __restrict__ centers,
    const float* __restrict__ svr_w,  const float* __restrict__ svr_b,
    const float* __restrict__ fc_w,   const float* __restrict__ fc_b,
    const float* __restrict__ out_w,  const float* __restrict__ out_b,
    float* __restrict__ out) {
  __shared__ __align__(16) float cbuf[2][2][CB_ELEMS];   // [buffer][head-slot]
  __shared__ float hid[ROWS_PER_BLK * HID_STRIDE];
  __shared__ float x2s[ROWS_PER_BLK];
  __shared__ float owpart[2][ROWS_PER_BLK];

  const int tid  = threadIdx.x;
  const int wave = tid >> 5;
  const int lane = tid & 31;
  const int half = lane >> 4;             // 0: lanes 0-15, 1: lanes 16-31
  const int l16  = lane & 15;
  const int mt   = wave & (MT_PER_BLK - 1);
  const int hg   = wave >> 2;             // head group (0/1)
  const int row0 = blockIdx.x * ROWS_PER_BLK;
  const int myrow = mt * 16 + l16;

  const unsigned long long centers_g = (unsigned long long)(uintptr_t)centers;
  const uint32_t cb_lds0 = (uint32_t)(uintptr_t)&cbuf[0][0][0];
  const uint32_t cb_lds1 = (uint32_t)(uintptr_t)&cbuf[1][0][0];

  // ||x_row||^2 for our 64 rows
  if (tid < ROWS_PER_BLK) {
    const float* p = x + (size_t)(row0 + tid) * D_DIM;
    float s = 0.0f;
#pragma unroll 16
    for (int d = 0; d < D_DIM; ++d) s += p[d] * p[d];
    x2s[tid] = s;
  }

  // A fragments of this wave's 16x64 x-tile (WMMA f32 16x16x4 A layout):
  //   lane l<16:  M=l, K=4c+0,4c+1 ; lane l>=16: M=l-16, K=4c+2,4c+3
  v2f xa[16];
  {
    const float* xp = x + (size_t)(row0 + myrow) * D_DIM + half * 2;
#pragma unroll
    for (int c = 0; c < 16; ++c)
      xa[c] = *reinterpret_cast<const v2f*>(xp + c * 4);
  }

  // Prologue: stage heads {0,1} into buffer 0.
  stage_pair(centers_g, 0, cb_lds0, tid);
  __syncthreads();   // x2s ready

  float x2r[8];
#pragma unroll
  for (int r = 0; r < 8; ++r) x2r[r] = x2s[mt * 16 + r + half * 8];

  // ---------------- Phase 1: 128 staged head-pairs ----------------
  for (int it = 0; it < H0_DIM / 2; ++it) {
    const int cur = it & 1;
    wait_async0();       // this wave's share of staged pair `it` has landed
    __syncthreads();     // all waves' shares landed; all done reading other buffer
    if (it + 1 < H0_DIM / 2)
      stage_pair(centers_g, it + 1, cur ? cb_lds0 : cb_lds1, tid);

    const int h = 2 * it + hg;
    const float* bufp = &cbuf[cur][hg][0];

    float racc[8];
#pragma unroll
    for (int r = 0; r < 8; ++r) racc[r] = 0.0f;

#pragma unroll
    for (int nt = 0; nt < 4; ++nt) {        // 4 tiles of 16 centers (50 padded)
      const int kc     = nt * 16 + l16;
      const bool valid = (kc < K_CENT);
      const int kcc    = valid ? kc : 0;

      const float* bp = bufp + kcc * CB_STRIDE + half * 2;
      const v2f zz = {};
      v2f bf[16];
      float c2p = 0.0f;                     // this half's share of ||center||^2
#pragma unroll
      for (int c = 0; c < 16; ++c) {
        v2f v = *reinterpret_cast<const v2f*>(bp + c * 4);
        v = valid ? v : zz;
        bf[c] = v;
        c2p += v.x * v.x + v.y * v.y;
      }
      const float c2v = c2p + __shfl_xor(c2p, 16);   // combine d-halves

      v8f acc = {};
#pragma unroll
      for (int c = 0; c < 16; ++c)
        acc = __builtin_amdgcn_wmma_f32_16x16x4_f32(
            false, xa[c], false, bf[c], (short)0, acc, false, false);

      const float w = valid ? svr_w[h * K_CENT + kcc] : 0.0f;
#pragma unroll
      for (int r = 0; r < 8; ++r) {
        // exp(-g*(x2 + c2 - 2*xc)) = exp(0.2*xc - 0.1*(x2+c2))
        float e = __expf(2.0f * GAMMA * acc[r] - GAMMA * (x2r[r] + c2v));
        racc[r] += e * w;
      }
    }

    // reduce over the 16 center-columns (within each 16-lane half)
#pragma unroll
    for (int r = 0; r < 8; ++r) {
      float v = racc[r];
      v += __shfl_xor(v, 1);
      v += __shfl_xor(v, 2);
      v += __shfl_xor(v, 4);
      v += __shfl_xor(v, 8);
      racc[r] = v;
    }
    if (l16 == 0) {
      const float bb = svr_b[h];
#pragma unroll
      for (int r = 0; r < 8; ++r)
        hid[(mt * 16 + r + half * 8) * HID_STRIDE + h] = tanhf(racc[r] + bb);
    }
  }
  __syncthreads();   // hid columns written by both head groups

  // ---------------- Phase 2: fc (256x256) + out ----------------
  float oacc[8];
#pragma unroll
  for (int r = 0; r < 8; ++r) oacc[r] = 0.0f;

  for (int t = 0; t < 8; ++t) {             // this wave's 8 column-tiles
    const int col = (hg * 8 + t) * 16 + l16;
    v8f acc = {};
    const float* ap = hid + myrow * HID_STRIDE + half * 2;        // A from LDS
    const float* bp = fc_w + (size_t)col * H0_DIM + half * 2;     // B from fc_w
#pragma unroll 8
    for (int c = 0; c < H0_DIM / 4; ++c) {  // 64 chained WMMAs, K = 256
      v2f a = *reinterpret_cast<const v2f*>(ap + c * 4);
      v2f b = *reinterpret_cast<const v2f*>(bp + c * 4);
      acc = __builtin_amdgcn_wmma_f32_16x16x4_f32(
          false, a, false, b, (short)0, acc, false, false);
    }
    const float fb = fc_b[col];
    const float ow = out_w[col];
#pragma unroll
    for (int r = 0; r < 8; ++r)
      oacc[r] += tanhf(acc[r] + fb) * ow;
  }

#pragma unroll
  for (int r = 0; r < 8; ++r) {
    float v = oacc[r];
    v += __shfl_xor(v, 1);
    v += __shfl_xor(v, 2);
    v += __shfl_xor(v, 4);
    v += __shfl_xor(v, 8);
    oacc[r] = v;
  }
  if (l16 == 0) {
#pragma unroll
    for (int r = 0; r < 8; ++r) owpart[hg][mt * 16 + r + half * 8] = oacc[r];
  }
  __syncthreads();
  if (tid < ROWS_PER_BLK)
    out[row0 + tid] = owpart[0][tid] + owpart[1][tid] + out_b[0];
}

// ---------------------------------------------------------------------------
extern "C" void kernel_launch(void* const* d_in, const int* in_sizes, int n_in,
                              void* d_out, int out_size, void* d_ws, size_t ws_size,
                              hipStream_t stream) {
  (void)in_sizes; (void)n_in; (void)out_size; (void)d_ws; (void)ws_size;
  const float* x       = (const float*)d_in[0];
  const float* centers = (const float*)d_in[1];
  const float* svr_w   = (const float*)d_in[2];
  const float* svr_b   = (const float*)d_in[3];
  const float* fc_w    = (const float*)d_in[4];
  const float* fc_b    = (const float*)d_in[5];
  const float* out_w   = (const float*)d_in[6];
  const float* out_b   = (const float*)d_in[7];
  float* out = (float*)d_out;

  const int nBlocks = 16384 / ROWS_PER_BLK;   // 256 blocks, 8 waves each
  kstonet_fused_kernel<<<nBlocks, 256, 0, stream>>>(
      x, centers, svr_w, svr_b, fc_w, fc_b, out_w, out_b, out);
}